// SparseAttention_60756607369846
// MI455X (gfx1250) — compile-verified
//
#include <hip/hip_runtime.h>

// CDNA5 wave32 WMMA types
typedef float v2f __attribute__((ext_vector_type(2)));
typedef float v8f __attribute__((ext_vector_type(8)));

constexpr int B_    = 2;
constexpr int NQ    = 8192;
constexpr int NKEY  = 16384;
constexpr int DIMQ  = 3;
constexpr int HC    = 64;
constexpr int CV    = 64;
constexpr int TOPK  = 16;
constexpr int QTILE = 16;    // queries per wave (WMMA M)
constexpr int KTILE = 16;    // keys per WMMA tile (WMMA N)
constexpr int UNROLL = 4;    // key tiles per loop iteration (4 WMMAs in flight)
constexpr int WAVES_PER_BLOCK = 8;
constexpr int QT_PER_BLOCK   = 4;   // query tiles per block
constexpr int KSPLIT         = 2;   // waves splitting the key range per query tile
constexpr int KEYS_PER_WAVE  = NKEY / KSPLIT;       // 8192
constexpr int COLSTRIDE = 20;  // padded LDS column stride (floats): bank-conflict-free
constexpr int NLISTS = 2 * KSPLIT;  // per-query candidate lists (2 lane-halves x 2 waves)
constexpr int NCAND  = UNROLL * 8;  // candidates per lane per iteration

__device__ __forceinline__ void topk_insert(float (&tval)[TOPK], unsigned (&tidx)[TOPK],
                                            float& worst, float s, unsigned id)
{
    if (s < worst) {                       // rare: gated by per-iteration min-tree too
        bool done = false;
        #pragma unroll
        for (int t = 0; t < TOPK; ++t) {
            bool hit = (!done) && (tval[t] == worst);
            tval[t] = hit ? s  : tval[t];
            tidx[t] = hit ? id : tidx[t];
            done = done || hit;
        }
        float wm = tval[0];
        #pragma unroll
        for (int t = 1; t < TOPK; ++t) wm = fmaxf(wm, tval[t]);
        worst = wm;
    }
}

extern "C" __global__ __launch_bounds__(32 * WAVES_PER_BLOCK)
void sparse_knn_attn_kernel(const float* __restrict__ qg, const float* __restrict__ kg,
                            const float* __restrict__ vg, const float* __restrict__ Wq,
                            const float* __restrict__ bqv, const float* __restrict__ Wk,
                            const float* __restrict__ bkv, float* __restrict__ outg)
{
    // Quad-buffered transpose tiles (per wave), cross-wave merge & final index lists.
    __shared__ __align__(16) float tileBuf[WAVES_PER_BLOCK][UNROLL][KTILE * COLSTRIDE];
    __shared__ float    mergeVal[QT_PER_BLOCK][QTILE][NLISTS * TOPK];
    __shared__ unsigned mergeIdx[QT_PER_BLOCK][QTILE][NLISTS * TOPK];
    __shared__ unsigned finalIdx[QT_PER_BLOCK][QTILE][TOPK];

    const int lane      = threadIdx.x & 31;
    const int wave      = threadIdx.x >> 5;
    const int tileInBlk = wave & (QT_PER_BLOCK - 1);   // which query tile of this block
    const int keyHalf   = wave >> 2;                   // which half of the key range
    const int gtile     = blockIdx.x * QT_PER_BLOCK + tileInBlk;   // 0..1023
    const int batch     = gtile / (NQ / QTILE);
    const int qt        = gtile % (NQ / QTILE);
    const int qbase     = qt * QTILE;
    const int col       = lane & 15;   // WMMA A row / D column index for this lane
    const int half      = lane >> 4;   // lane half (K-split for A/B, row-half for D)

    // ---- Fold both Linear layers into a 3x3 bilinear form:
    // (Wq q + bq) . (Wk k + bk) = q^T M k + u.k + w.q + cc
    float M00=0,M01=0,M02=0,M10=0,M11=0,M12=0,M20=0,M21=0,M22=0;
    float u0=0,u1=0,u2=0, w0=0,w1=0,w2=0, cc=0;
    #pragma unroll 4
    for (int h = 0; h < HC; ++h) {
        float a0=Wq[h*3+0], a1=Wq[h*3+1], a2=Wq[h*3+2];
        float b0=Wk[h*3+0], b1=Wk[h*3+1], b2=Wk[h*3+2];
        float pq=bqv[h], pk=bkv[h];
        M00 += a0*b0; M01 += a0*b1; M02 += a0*b2;
        M10 += a1*b0; M11 += a1*b1; M12 += a1*b2;
        M20 += a2*b0; M21 += a2*b1; M22 += a2*b2;
        u0 += pq*b0; u1 += pq*b1; u2 += pq*b2;
        w0 += a0*pk; w1 += a1*pk; w2 += a2*pk;
        cc += pq*pk;
    }

    // ---- Phase 1: WMMA distance-proxy tiles + streaming per-lane top-16 ----
    // A (16x4 f32, 2 VGPR): lanes 0-15 hold K0/K1, lanes 16-31 hold K2/K3; row M = lane%16.
    // Bake -2 into A so D = |k|^2 - 2 q.k  (monotone in Euclidean distance per query).
    const float* qrow = qg + ((size_t)batch * NQ + qbase + col) * DIMQ;
    float qx = qrow[0], qy = qrow[1], qz = qrow[2];
    v2f A;
    A.x = half ? -2.0f * qz : -2.0f * qx;
    A.y = half ?  0.0f      : -2.0f * qy;

    float    tval[TOPK];
    unsigned tidx[TOPK];
    #pragma unroll
    for (int t = 0; t < TOPK; ++t) { tval[t] = __builtin_inff(); tidx[t] = 0u; }
    float worst = __builtin_inff();

    const int kstart = keyHalf * KEYS_PER_WAVE;
    const int kend   = kstart + KEYS_PER_WAVE;

    for (int kb = kstart; kb < kend; kb += UNROLL * KTILE) {
        // UNROLL independent key sub-tiles: overlap global loads + WMMAs + LDS stores.
        #pragma unroll
        for (int u = 0; u < UNROLL; ++u) {
            const float* kr = kg + ((size_t)batch * NKEY + kb + u * KTILE + col) * DIMQ;
            float kx = kr[0], ky = kr[1], kz = kr[2];
            float ksq = kx*kx + ky*ky + kz*kz;
            // B (4x16 f32, 2 VGPR): VGPR0 = rows K0 (lanes 0-15) / K1 (lanes 16-31),
            //                       VGPR1 = rows K2 (lanes 0-15) / K3=0 (lanes 16-31).
            v2f Bm;
            Bm.x = half ? ky   : kx;
            Bm.y = half ? 0.0f : kz;
            v8f C;
            #pragma unroll
            for (int t = 0; t < 8; ++t) C[t] = ksq;  // per-column |k|^2 rides the WMMA

            C = __builtin_amdgcn_wmma_f32_16x16x4_f32(false, A, false, Bm, (short)0, C,
                                                      false, false);
            // Scatter D tile to LDS as [column][row], padded stride (conflict-free).
            float4* st = (float4*)&tileBuf[wave][u][col * COLSTRIDE + half * 8];
            st[0] = make_float4(C[0], C[1], C[2], C[3]);
            st[1] = make_float4(C[4], C[5], C[6], C[7]);
        }
        __asm__ volatile("s_wait_dscnt 0" ::: "memory");

        // Straight-line transposed read-back of ALL candidates first (loads overlap),
        // all for this lane's query `col`, columns half*8 .. half*8+7 of each sub-tile.
        float cand[NCAND];
        #pragma unroll
        for (int u = 0; u < UNROLL; ++u)
            #pragma unroll
            for (int j = 0; j < 8; ++j)
                cand[u * 8 + j] = tileBuf[wave][u][(half * 8 + j) * COLSTRIDE + col];

        // Pairwise min-tree: single test to skip the whole insert block.
        float m16[16];
        #pragma unroll
        for (int t = 0; t < 16; ++t) m16[t] = fminf(cand[t], cand[t + 16]);
        float m8[8];
        #pragma unroll
        for (int t = 0; t < 8; ++t) m8[t] = fminf(m16[t], m16[t + 8]);
        float m4[4];
        #pragma unroll
        for (int t = 0; t < 4; ++t) m4[t] = fminf(m8[t], m8[t + 4]);
        float cmin = fminf(fminf(m4[0], m4[1]), fminf(m4[2], m4[3]));

        if (cmin < worst) {                       // ~1/3 of iterations after warm-up
            #pragma unroll
            for (int u = 0; u < UNROLL; ++u)
                #pragma unroll
                for (int j = 0; j < 8; ++j)
                    topk_insert(tval, tidx, worst, cand[u * 8 + j],
                                (unsigned)(kb + u * KTILE + half * 8 + j));
        }
        __asm__ volatile("s_wait_dscnt 0" ::: "memory");   // WAR guard for next iter
    }

    // ---- Cross-wave merge: 4 candidate lists per query -> final top-16 ----
    const int listId = keyHalf * 2 + half;
    #pragma unroll
    for (int t = 0; t < TOPK; ++t) {
        mergeVal[tileInBlk][col][listId * TOPK + t] = tval[t];
        mergeIdx[tileInBlk][col][listId * TOPK + t] = tidx[t];
    }
    __syncthreads();

    if (threadIdx.x < QT_PER_BLOCK * QTILE) {          // 64 selector lanes
        const int stile = threadIdx.x >> 4;
        const int sq    = threadIdx.x & 15;
        #pragma unroll 1
        for (int slot = 0; slot < TOPK; ++slot) {
            float best = __builtin_inff(); int bpos = 0;
            #pragma unroll 1
            for (int t = 0; t < NLISTS * TOPK; ++t) {
                float vv = mergeVal[stile][sq][t];
                if (vv < best) { best = vv; bpos = t; }
            }
            finalIdx[stile][sq][slot] = mergeIdx[stile][sq][bpos];
            mergeVal[stile][sq][bpos] = __builtin_inff();
        }
    }
    __syncthreads();

    // ---- Phase 2: bilinear scores, softmax(16), weighted v-gather ----
    // All 256 threads: task = (query tile, query, 16-float output quarter).
    const int task   = threadIdx.x;
    const int ptile  = task >> 6;
    const int pq     = (task >> 2) & 15;
    const int pquart = task & 3;
    const int gt2    = blockIdx.x * QT_PER_BLOCK + ptile;
    const int b2     = gt2 / (NQ / QTILE);
    const int qb2    = (gt2 % (NQ / QTILE)) * QTILE;
    const size_t gq  = (size_t)b2 * NQ + qb2 + pq;

    const float* qp2 = qg + gq * DIMQ;
    float q0 = qp2[0], q1 = qp2[1], q2 = qp2[2];
    float qterm = w0*q0 + w1*q1 + w2*q2 + cc;

    unsigned nid[TOPK];
    float    sc[TOPK];
    #pragma unroll
    for (int j = 0; j < TOPK; ++j) {
        unsigned i = finalIdx[ptile][pq][j];
        nid[j] = i;
        const float* kp = kg + ((size_t)b2 * NKEY + i) * DIMQ;
        float k0 = kp[0], k1 = kp[1], k2 = kp[2];
        sc[j] = q0 * (M00*k0 + M01*k1 + M02*k2)
              + q1 * (M10*k0 + M11*k1 + M12*k2)
              + q2 * (M20*k0 + M21*k1 + M22*k2)
              + u0*k0 + u1*k1 + u2*k2 + qterm;
    }
    float mx = sc[0];
    #pragma unroll
    for (int j = 1; j < TOPK; ++j) mx = fmaxf(mx, sc[j]);
    float ssum = 0.0f;
    #pragma unroll
    for (int j = 0; j < TOPK; ++j) { sc[j] = __expf(sc[j] - mx); ssum += sc[j]; }
    float inv = 1.0f / ssum;

    float acc[16];
    #pragma unroll
    for (int c = 0; c < 16; ++c) acc[c] = 0.0f;
    #pragma unroll 2
    for (int j = 0; j < TOPK; ++j) {
        float p = sc[j] * inv;
        const float4* vp = (const float4*)(vg + ((size_t)b2 * NKEY + nid[j]) * CV
                                           + pquart * 16);
        #pragma unroll
        for (int t = 0; t < 4; ++t) {
            float4 vv = vp[t];
            acc[4*t+0] += p * vv.x;  acc[4*t+1] += p * vv.y;
            acc[4*t+2] += p * vv.z;  acc[4*t+3] += p * vv.w;
        }
    }
    float4* op = (float4*)(outg + gq * CV + pquart * 16);
    #pragma unroll
    for (int t = 0; t < 4; ++t)
        op[t] = make_float4(acc[4*t+0], acc[4*t+1], acc[4*t+2], acc[4*t+3]);
}

extern "C" void kernel_launch(void* const* d_in, const int* in_sizes, int n_in,
                              void* d_out, int out_size, void* d_ws, size_t ws_size,
                              hipStream_t stream) {
    (void)in_sizes; (void)n_in; (void)d_ws; (void)ws_size; (void)out_size;
    const float* q  = (const float*)d_in[0];
    const float* k  = (const float*)d_in[1];
    const float* v  = (const float*)d_in[2];
    const float* Wq = (const float*)d_in[3];
    const float* bq = (const float*)d_in[4];
    const float* Wk = (const float*)d_in[5];
    const float* bk = (const float*)d_in[6];
    // d_in[7] = top_k (16), d_in[8] = chunk_size (1024): fixed by construction.

    const int totalTiles = B_ * (NQ / QTILE);              // 1024 query tiles
    dim3 grid(totalTiles / QT_PER_BLOCK);                  // 256 blocks
    dim3 block(32 * WAVES_PER_BLOCK);                      // 256 threads = 8 waves
    hipLaunchKernelGGL(sparse_knn_attn_kernel, grid, block, 0, stream,
                       q, k, v, Wq, bq, Wk, bk, (float*)d_out);
}